// SolvGNNV5_51462298141155
// MI455X (gfx1250) — compile-verified
//
#include <hip/hip_runtime.h>
#include <hip/hip_bf16.h>

// ---------------------------------------------------------------------------
// SolvGNN forward on gfx1250 (MI455X, wave32).
// GEMMs use V_WMMA_F32_16X16X4_F32 (fp32 matrix pipe), 4 independent
// accumulator chains per wave (16x64 strip) to hide WMMA latency.
// Edge aggregation uses L2-resident global f32 atomics (x/agg = 25.6MB each,
// global L2 = 192MB, so the 410MB/layer gather+scatter never touches HBM).
// ---------------------------------------------------------------------------

#define N_NODES  50000
#define N_EDGES  800000
#define N_GRAPHS 500
#define IN_DIM   74
#define HIDDEN   128
#define N_LAYERS 25

typedef float v2f __attribute__((ext_vector_type(2)));
typedef float v8f __attribute__((ext_vector_type(8)));

// ---- pad h [N x 74] -> x [N x 128] (cols 74..127 = 0) ----------------------
__global__ void pad_h_kernel(const float* __restrict__ h, float* __restrict__ x, int n) {
    int i = blockIdx.x * blockDim.x + threadIdx.x;       // n * 128 threads
    if (i >= n * HIDDEN) return;
    int row = i >> 7, c = i & 127;
    x[i] = (c < IN_DIM) ? h[row * IN_DIM + c] : 0.0f;
}

// ---- pad W0 [74 x 128] -> [128 x 128] (rows 74..127 = 0) -------------------
__global__ void pad_w0_kernel(const float* __restrict__ w, float* __restrict__ wp) {
    int i = blockIdx.x * blockDim.x + threadIdx.x;       // 128*128 threads
    if (i >= HIDDEN * HIDDEN) return;
    int r = i >> 7;
    wp[i] = (r < IN_DIM) ? w[i] : 0.0f;                  // same flat index while r<74
}

// ---- vectorized copy (agg := x), count in float4 units ---------------------
__global__ void copy_f4_kernel(const float4* __restrict__ src4, float4* __restrict__ dst4, int n4) {
    int i = blockIdx.x * blockDim.x + threadIdx.x;
    if (i < n4) dst4[i] = src4[i];
}

// ---- zero fill -------------------------------------------------------------
__global__ void zero_kernel(float* __restrict__ p, int n) {
    int i = blockIdx.x * blockDim.x + threadIdx.x;
    if (i < n) p[i] = 0.0f;
}

// ---- edge scatter: agg[dst] += x[src]; one thread per (edge, 4 channels) ---
__global__ void edge_scatter_kernel(const int* __restrict__ src, const int* __restrict__ dst,
                                    const float* __restrict__ x, float* __restrict__ agg,
                                    int nEdges) {
    int gid = blockIdx.x * blockDim.x + threadIdx.x;     // nEdges * 32 threads
    int e = gid >> 5;                                    // 128 ch / 4 = 32 groups
    int g = gid & 31;
    if (e >= nEdges) return;
    int s = src[e], d = dst[e];
    const float4 v = ((const float4*)(x + (size_t)s * HIDDEN))[g];
    float* out = agg + (size_t)d * HIDDEN + (g << 2);
    atomicAdd(out + 0, v.x);
    atomicAdd(out + 1, v.y);
    atomicAdd(out + 2, v.z);
    atomicAdd(out + 3, v.w);
}

// ---- fp32 WMMA GEMM: C[M,N] = act(A[M,K] @ W[K,N] + bias) ------------------
// One wave per 16x64 output strip = 4 independent 16x16 WMMA accumulator
// chains (hides WMMA latency, 4x A-fragment reuse). 8 waves per block.
// EXEC stays all-ones through the WMMA loop (wave-uniform early-out, clamped
// A-row addressing; clamped garbage rows are simply never stored).
// A fragment (16x4 f32, 2 VGPRs): lane m = lane&15; vgpr0/1 hold K{0,1} for
// lanes 0-15, K{2,3} for lanes 16-31. B (4x16): K-rows striped across lanes.
// C/D (16x16, 8 VGPRs): vgpr r -> row r (lanes 0-15) / r+8 (lanes 16-31).
// Requires: N % 64 == 0, K % 4 == 0.
__global__ __launch_bounds__(256, 2)
void gemm_wmma_f32_kernel(const float* __restrict__ A, const float* __restrict__ W,
                          const float* __restrict__ bias, float* __restrict__ C,
                          int M, int K, int N, int relu) {
    const int lane   = threadIdx.x & 31;
    const int wave   = threadIdx.x >> 5;
    const int nGrp   = N >> 6;                           // 64-column groups
    const int tilesM = (M + 15) >> 4;
    int strip = blockIdx.x * 8 + wave;
    if (strip >= tilesM * nGrp) return;                  // wave-uniform branch
    int tm = strip / nGrp;
    int tg = strip - tm * nGrp;
    int m0 = tm << 4, n0 = tg << 6;

    int am = m0 + (lane & 15);
    if (am >= M) am = M - 1;                             // clamp; rows >= M never stored
    const int koff = (lane & 16) ? 2 : 0;

    // Pointer-stepped addressing: no per-iteration 64-bit address recompute.
    const v2f*   Ap = (const v2f*)(A + (size_t)am * K) + (koff >> 1);
    const float* Bp = W + (size_t)koff * N + n0 + (lane & 15);
    const size_t sB = (size_t)N;                         // floats between K rows

    v8f acc0 = {}, acc1 = {}, acc2 = {}, acc3 = {};
#pragma unroll 8
    for (int k = 0; k < K; k += 4) {
        if ((k & 31) == 0)                               // constant-folded under unroll
            __builtin_prefetch(Bp + 8 * sB, 0, 1);       // global_prefetch_b8, next chunk
        v2f a = *Ap; Ap += 2;                            // global_load_b64
        v2f b0, b1, b2, b3;
        b0.x = Bp[0];       b0.y = Bp[sB];
        b1.x = Bp[16];      b1.y = Bp[sB + 16];
        b2.x = Bp[32];      b2.y = Bp[sB + 32];
        b3.x = Bp[48];      b3.y = Bp[sB + 48];
        Bp += 4 * sB;
        acc0 = __builtin_amdgcn_wmma_f32_16x16x4_f32(false, a, false, b0, (short)0, acc0, false, false);
        acc1 = __builtin_amdgcn_wmma_f32_16x16x4_f32(false, a, false, b1, (short)0, acc1, false, false);
        acc2 = __builtin_amdgcn_wmma_f32_16x16x4_f32(false, a, false, b2, (short)0, acc2, false, false);
        acc3 = __builtin_amdgcn_wmma_f32_16x16x4_f32(false, a, false, b3, (short)0, acc3, false, false);
    }

    const int rbase = m0 + ((lane & 16) ? 8 : 0);
    const int cn    = n0 + (lane & 15);
    v8f accs[4] = {acc0, acc1, acc2, acc3};
#pragma unroll
    for (int t = 0; t < 4; ++t) {
        int col = cn + 16 * t;
        float bv = bias[col];
#pragma unroll
        for (int r = 0; r < 8; ++r) {
            int row = rbase + r;
            if (row < M) {
                float v = accs[t][r] + bv;
                if (relu) v = fmaxf(v, 0.0f);
                C[(size_t)row * N + col] = v;
            }
        }
    }
}

// ---- per-graph mean pooling ------------------------------------------------
__global__ void pool_accum_kernel(const float* __restrict__ x, const int* __restrict__ gid,
                                  float* __restrict__ pooled, float* __restrict__ counts, int n) {
    int i = blockIdx.x * blockDim.x + threadIdx.x;       // n * 128 threads
    if (i >= n * HIDDEN) return;
    int row = i >> 7, c = i & 127;
    int g = gid[row];
    atomicAdd(&pooled[g * HIDDEN + c], x[i]);
    if (c == 0) atomicAdd(&counts[g], 1.0f);
}

__global__ void pool_div_kernel(float* __restrict__ pooled, const float* __restrict__ counts) {
    int i = blockIdx.x * blockDim.x + threadIdx.x;       // N_GRAPHS*128 threads
    if (i >= N_GRAPHS * HIDDEN) return;
    pooled[i] /= counts[i >> 7];
}

// ---- BatchNorm (batch stats over G rows) + LeakyReLU, one block per column -
__global__ void bn_leaky_kernel(float* __restrict__ y, const float* __restrict__ gam,
                                const float* __restrict__ bet, int G, int D) {
    const int j = blockIdx.x;
    __shared__ float s1[256], s2[256];
    float s = 0.0f, q = 0.0f;
    for (int i = threadIdx.x; i < G; i += blockDim.x) {
        float v = y[(size_t)i * D + j];
        s += v; q += v * v;
    }
    s1[threadIdx.x] = s; s2[threadIdx.x] = q;
    __syncthreads();
    for (int off = blockDim.x >> 1; off > 0; off >>= 1) {
        if ((int)threadIdx.x < off) {
            s1[threadIdx.x] += s1[threadIdx.x + off];
            s2[threadIdx.x] += s2[threadIdx.x + off];
        }
        __syncthreads();
    }
    float mu  = s1[0] / (float)G;
    float var = s2[0] / (float)G - mu * mu;              // biased variance
    float inv = rsqrtf(var + 1e-5f);
    float gg = gam[j], bb = bet[j];
    for (int i = threadIdx.x; i < G; i += blockDim.x) {
        size_t idx = (size_t)i * D + j;
        float v = (y[idx] - mu) * inv * gg + bb;
        y[idx] = v > 0.0f ? v : 0.01f * v;
    }
}

// ---- final [G x K] @ [K x 1] + b, one wave per row -------------------------
__global__ void final_dot_kernel(const float* __restrict__ y, const float* __restrict__ w,
                                 const float* __restrict__ b, float* __restrict__ out,
                                 int G, int K) {
    int row  = blockIdx.x * (blockDim.x >> 5) + (threadIdx.x >> 5);
    int lane = threadIdx.x & 31;
    if (row >= G) return;
    float s = 0.0f;
    for (int k = lane; k < K; k += 32) s += y[(size_t)row * K + k] * w[k];
    for (int off = 16; off > 0; off >>= 1) s += __shfl_down(s, off, 32);
    if (lane == 0) out[row] = s + b[0];
}

// ---------------------------------------------------------------------------
extern "C" void kernel_launch(void* const* d_in, const int* in_sizes, int n_in,
                              void* d_out, int out_size, void* d_ws, size_t ws_size,
                              hipStream_t stream) {
    (void)in_sizes; (void)n_in; (void)out_size; (void)ws_size;

    // Input order: h, src, dst, graph_id, gin_w[25], gin_b[25],
    //              mlp_w[4], mlp_b[4], bn_g[3], bn_b[3]
    const float* h   = (const float*)d_in[0];
    const int*   src = (const int*)  d_in[1];
    const int*   dst = (const int*)  d_in[2];
    const int*   gid = (const int*)  d_in[3];
    const float* gin_w[N_LAYERS]; const float* gin_b[N_LAYERS];
    for (int i = 0; i < N_LAYERS; ++i) {
        gin_w[i] = (const float*)d_in[4 + i];
        gin_b[i] = (const float*)d_in[4 + N_LAYERS + i];
    }
    const float* mlp_w[4]; const float* mlp_b[4];
    for (int i = 0; i < 4; ++i) {
        mlp_w[i] = (const float*)d_in[4 + 2 * N_LAYERS + i];
        mlp_b[i] = (const float*)d_in[4 + 2 * N_LAYERS + 4 + i];
    }
    const float* bn_g[3]; const float* bn_b[3];
    for (int i = 0; i < 3; ++i) {
        bn_g[i] = (const float*)d_in[4 + 2 * N_LAYERS + 8 + i];
        bn_b[i] = (const float*)d_in[4 + 2 * N_LAYERS + 11 + i];
    }
    float* out = (float*)d_out;

    // Workspace carve-out (256B aligned)
    char* ws = (char*)d_ws;
    auto carve = [&](size_t bytes) -> float* {
        float* p = (float*)ws;
        ws += (bytes + 255) & ~(size_t)255;
        return p;
    };
    const size_t NODE_F = (size_t)N_NODES * HIDDEN;
    float* x      = carve(NODE_F * 4);                  // node features (ping)
    float* agg    = carve(NODE_F * 4);                  // x + neighbor sum
    float* xn     = carve(NODE_F * 4);                  // node features (pong)
    float* w0p    = carve((size_t)HIDDEN * HIDDEN * 4); // padded W0
    float* pooled = carve((size_t)N_GRAPHS * HIDDEN * 4);
    float* counts = carve((size_t)N_GRAPHS * 4);
    float* y1     = carve((size_t)N_GRAPHS * 2048 * 4);
    float* y2     = carve((size_t)N_GRAPHS * 1024 * 4);
    float* y3     = carve((size_t)N_GRAPHS * 512 * 4);

    const int T = 256;

    // Stage 0: pad inputs
    pad_h_kernel<<<(N_NODES * HIDDEN + T - 1) / T, T, 0, stream>>>(h, x, N_NODES);
    pad_w0_kernel<<<(HIDDEN * HIDDEN + T - 1) / T, T, 0, stream>>>(gin_w[0], w0p);

    // Stage 1: 25 GIN layers
    const int n4         = (int)(NODE_F / 4);
    const int edgeThr    = N_EDGES * 32;                 // 32 float4-groups per edge
    const int ginStrips  = ((N_NODES + 15) / 16) * (HIDDEN / 64);
    const int ginBlocks  = (ginStrips + 7) / 8;
    for (int l = 0; l < N_LAYERS; ++l) {
        copy_f4_kernel<<<(n4 + T - 1) / T, T, 0, stream>>>((const float4*)x, (float4*)agg, n4);
        edge_scatter_kernel<<<(edgeThr + T - 1) / T, T, 0, stream>>>(src, dst, x, agg, N_EDGES);
        const float* W = (l == 0) ? w0p : gin_w[l];
        gemm_wmma_f32_kernel<<<ginBlocks, T, 0, stream>>>(agg, W, gin_b[l], xn,
                                                          N_NODES, HIDDEN, HIDDEN, /*relu=*/1);
        float* t = x; x = xn; xn = t;                    // ping-pong
    }

    // Stage 2: per-graph mean pooling
    zero_kernel<<<(N_GRAPHS * HIDDEN + T - 1) / T, T, 0, stream>>>(pooled, N_GRAPHS * HIDDEN);
    zero_kernel<<<(N_GRAPHS + T - 1) / T, T, 0, stream>>>(counts, N_GRAPHS);
    pool_accum_kernel<<<(N_NODES * HIDDEN + T - 1) / T, T, 0, stream>>>(x, gid, pooled, counts, N_NODES);
    pool_div_kernel<<<(N_GRAPHS * HIDDEN + T - 1) / T, T, 0, stream>>>(pooled, counts);

    // Stage 3: MLP head (Linear -> BN(batch stats) -> LeakyReLU) x3, then Linear->1
    auto mlp_gemm = [&](const float* Ain, const float* Wm, const float* bm, float* Cout,
                        int M, int K, int N) {
        int strips = ((M + 15) / 16) * (N / 64);
        gemm_wmma_f32_kernel<<<(strips + 7) / 8, T, 0, stream>>>(Ain, Wm, bm, Cout, M, K, N, 0);
    };
    mlp_gemm(pooled, mlp_w[0], mlp_b[0], y1, N_GRAPHS, 128, 2048);
    bn_leaky_kernel<<<2048, T, 0, stream>>>(y1, bn_g[0], bn_b[0], N_GRAPHS, 2048);
    mlp_gemm(y1, mlp_w[1], mlp_b[1], y2, N_GRAPHS, 2048, 1024);
    bn_leaky_kernel<<<1024, T, 0, stream>>>(y2, bn_g[1], bn_b[1], N_GRAPHS, 1024);
    mlp_gemm(y2, mlp_w[2], mlp_b[2], y3, N_GRAPHS, 1024, 512);
    bn_leaky_kernel<<<512, T, 0, stream>>>(y3, bn_g[2], bn_b[2], N_GRAPHS, 512);

    final_dot_kernel<<<(N_GRAPHS * 32 + T - 1) / T, T, 0, stream>>>(y3, mlp_w[3], mlp_b[3],
                                                                    out, N_GRAPHS, 512);
}